// Max_19043884990479
// MI455X (gfx1250) — compile-verified
//
#include <hip/hip_runtime.h>
#include <stdint.h>

// Problem geometry from the reference: [B=128, 1, 512, 512] f32.
#define ROW_N     (512 * 512)            // elements per batch row
#define CHUNKS    8                      // workgroups per row
#define CHUNK_N   (ROW_N / CHUNKS)       // 32768 elements per block
#define TPB       256                    // 8 wave32s per block
#define F4_PER_TD (CHUNK_N / (TPB * 4))  // 32 float4 per thread
#define STAGES    4
#define STAGE_F4  (F4_PER_TD / STAGES)   // 8 float4 per thread per stage

// ---- CDNA5 async global<->LDS copy path (ASYNCcnt-tracked, ISA ch.10) ----
__device__ __forceinline__ void async_load_b128(unsigned lds_off, const void* gaddr) {
  // copies 16B per lane from the global address gaddr to the LDS offset lds_off, fire-and-forget (ASYNCcnt++)
  asm volatile("global_load_async_to_lds_b128 %0, %1, off"
               :: "v"(lds_off), "v"(gaddr) : "memory");
}
__device__ __forceinline__ void async_store_b128(void* gaddr, unsigned lds_off) {
  // copies 16B per lane from the LDS offset lds_off to the global address gaddr (ASYNCcnt++)
  asm volatile("global_store_async_from_lds_b128 %0, %1, off"
               :: "v"(gaddr), "v"(lds_off) : "memory");
}
__device__ __forceinline__ void wait_async0() {
  asm volatile("s_wait_asynccnt 0x0" ::: "memory");
}

// Branchless sorted top-3 insert (v0 >= v1 >= v2); strict '>' keeps the
// lowest index on ties, matching jax.lax.top_k. Lowers to v_cmp + dual
// v_cndmask chains, no divergence; fabs folds into compare src modifiers.
__device__ __forceinline__ void insert3(float a, int idx,
                                        float& v0, float& v1, float& v2,
                                        int& i0, int& i1, int& i2) {
  const bool gt0 = a > v0;
  const bool gt1 = a > v1;
  const bool gt2 = a > v2;
  v2 = gt1 ? v1 : (gt2 ? a   : v2);
  i2 = gt1 ? i1 : (gt2 ? idx : i2);
  v1 = gt0 ? v0 : (gt1 ? a   : v1);
  i1 = gt0 ? i0 : (gt1 ? idx : i1);
  v0 = gt0 ? a   : v0;
  i0 = gt0 ? idx : i0;
}

// Kernel 1: per-block top-3 of |difference| with the weight->out copy running
// entirely on the async engine (global -> LDS -> global), overlapped with the
// VALU scan. Exactly roofline traffic: 2x128MB read + 128MB write.
__global__ __launch_bounds__(TPB) void scan_and_copy(
    const float* __restrict__ diff, const float* __restrict__ w,
    float* __restrict__ out, float* __restrict__ pv, int* __restrict__ pi) {
  const int blk   = blockIdx.x;            // row * CHUNKS + chunk
  const int row   = blk / CHUNKS;
  const int chunk = blk - row * CHUNKS;
  const int tid   = threadIdx.x;

  const size_t base = (size_t)row * ROW_N + (size_t)chunk * CHUNK_N;
  const float4* __restrict__ d4 = (const float4*)(diff + base);
  const float4* __restrict__ w4 = (const float4*)(w + base);
  float4* __restrict__       o4 = (float4*)(out + base);

  // Ping-pong staging buffers for the async copy (2 x 32KB).
  __shared__ float4 cb[2][STAGE_F4 * TPB];
  const unsigned cb_base = (unsigned)(uintptr_t)(&cb[0][0]);  // LDS byte offset

  float v0 = -1.0f, v1 = -1.0f, v2 = -1.0f;   // |x| >= 0 always wins
  int   i0 = 0, i1 = 0, i2 = 0;
  const int chunkOff = chunk * CHUNK_N;

  for (int s = 0; s < STAGES; ++s) {
    const unsigned sb = (unsigned)(s & 1) * (STAGE_F4 * TPB * 16u);

    // Issue this stage's async loads: weight tile -> LDS (no VGPR data).
#pragma unroll
    for (int k = 0; k < STAGE_F4; ++k) {
      const int f4 = (s * STAGE_F4 + k) * TPB + tid;
      const unsigned lo = cb_base + sb + (unsigned)((k * TPB + tid) * 16);
      async_load_b128(lo, (const void*)(w4 + f4));
    }

    // Overlapped VALU work: scan this stage of |difference|.
#pragma unroll
    for (int k = 0; k < STAGE_F4; ++k) {
      const int f4 = (s * STAGE_F4 + k) * TPB + tid;
      // gfx1250 speculative prefetch one stage ahead (OOB-safe, ISA 10.5).
      __builtin_prefetch(d4 + f4 + STAGE_F4 * TPB, 0, 0);
      const float4 d = d4[f4];
      const int e = chunkOff + f4 * 4;
      insert3(fabsf(d.x), e + 0, v0, v1, v2, i0, i1, i2);
      insert3(fabsf(d.y), e + 1, v0, v1, v2, i0, i1, i2);
      insert3(fabsf(d.z), e + 2, v0, v1, v2, i0, i1, i2);
      insert3(fabsf(d.w), e + 3, v0, v1, v2, i0, i1, i2);
    }

    // Loads of stage s (and stores of stage s-1) retired -> LDS tile valid,
    // and the s+2 reuse of this ping-pong buffer is hazard-free.
    wait_async0();

    // Issue async stores: LDS tile -> out (completion covered by S_ENDPGM's
    // implicit wait-idle; reduction below uses a disjoint LDS region).
#pragma unroll
    for (int k = 0; k < STAGE_F4; ++k) {
      const int f4 = (s * STAGE_F4 + k) * TPB + tid;
      const unsigned lo = cb_base + sb + (unsigned)((k * TPB + tid) * 16);
      async_store_b128((void*)(o4 + f4), lo);
    }
  }

  // Block-level tree merge of per-thread top-3 lists through LDS.
  __shared__ float sv[TPB * 3];
  __shared__ int   si[TPB * 3];
  sv[tid * 3 + 0] = v0; sv[tid * 3 + 1] = v1; sv[tid * 3 + 2] = v2;
  si[tid * 3 + 0] = i0; si[tid * 3 + 1] = i1; si[tid * 3 + 2] = i2;
  __syncthreads();
  for (int s = TPB / 2; s > 0; s >>= 1) {
    if (tid < s) {
      const int p = (tid + s) * 3;
      insert3(sv[p + 0], si[p + 0], v0, v1, v2, i0, i1, i2);
      insert3(sv[p + 1], si[p + 1], v0, v1, v2, i0, i1, i2);
      insert3(sv[p + 2], si[p + 2], v0, v1, v2, i0, i1, i2);
      sv[tid * 3 + 0] = v0; sv[tid * 3 + 1] = v1; sv[tid * 3 + 2] = v2;
      si[tid * 3 + 0] = i0; si[tid * 3 + 1] = i1; si[tid * 3 + 2] = i2;
    }
    __syncthreads();
  }
  if (tid == 0) {
    const int o = blk * 3;
    pv[o + 0] = v0; pv[o + 1] = v1; pv[o + 2] = v2;
    pi[o + 0] = i0; pi[o + 1] = i1; pi[o + 2] = i2;
  }
}

// Kernel 2: one thread per row merges its CHUNKS partial top-3 lists and
// scatters the three +1.0 fixups. Epoch condition evaluated on-device so the
// launch is graph-capture-safe and deterministic; writes read weight (not
// out), so replays are idempotent.
__global__ void merge_and_fixup(const float* __restrict__ pv,
                                const int* __restrict__ pi,
                                const float* __restrict__ w,
                                float* __restrict__ out,
                                const int* __restrict__ epoch, int B) {
  const int row = blockIdx.x * blockDim.x + threadIdx.x;
  if (row >= B) return;
  const int e = epoch[0];
  const bool active = (e > 200) && (e < 1000) && (e % 20 == 0);
  if (!active) return;  // reference returns plain weight; kernel 1 wrote it

  float v0 = -1.0f, v1 = -1.0f, v2 = -1.0f;
  int   i0 = 0, i1 = 0, i2 = 0;
#pragma unroll
  for (int c = 0; c < CHUNKS; ++c) {
    const int o = (row * CHUNKS + c) * 3;
    insert3(pv[o + 0], pi[o + 0], v0, v1, v2, i0, i1, i2);
    insert3(pv[o + 1], pi[o + 1], v0, v1, v2, i0, i1, i2);
    insert3(pv[o + 2], pi[o + 2], v0, v1, v2, i0, i1, i2);
  }
  const size_t base = (size_t)row * ROW_N;
  out[base + i0] = w[base + i0] + 1.0f;
  out[base + i1] = w[base + i1] + 1.0f;
  out[base + i2] = w[base + i2] + 1.0f;
}

extern "C" void kernel_launch(void* const* d_in, const int* in_sizes, int n_in,
                              void* d_out, int out_size, void* d_ws, size_t ws_size,
                              hipStream_t stream) {
  (void)n_in; (void)out_size; (void)ws_size;
  const float* diff  = (const float*)d_in[0];
  const float* w     = (const float*)d_in[1];
  const int*   epoch = (const int*)d_in[2];
  float*       out   = (float*)d_out;

  const long long total = (long long)in_sizes[0];
  const int B = (int)(total / ROW_N);              // 128

  // Workspace layout: [B*CHUNKS*3 floats][B*CHUNKS*3 ints] (~24 KB)
  float* pv = (float*)d_ws;
  int*   pi = (int*)((char*)d_ws + (size_t)B * CHUNKS * 3 * sizeof(float));

  scan_and_copy<<<B * CHUNKS, TPB, 0, stream>>>(diff, w, out, pv, pi);
  merge_and_fixup<<<(B + 63) / 64, 64, 0, stream>>>(pv, pi, w, out, epoch, B);
}